// Dihedral2Coord_6975026888753
// MI455X (gfx1250) — compile-verified
//
#include <hip/hip_runtime.h>
#include <math.h>

// WMMA operand types for V_WMMA_F32_16X16X4_F32 (wave32):
//   A: 16x4 f32  -> 2 VGPRs ; B: 4x16 f32 -> 2 VGPRs ; C/D: 16x16 f32 -> 8 VGPRs
typedef float v2f __attribute__((ext_vector_type(2)));
typedef float v8f __attribute__((ext_vector_type(8)));

constexpr int KSTEPS  = 128;   // dihedral steps (scan length)
constexpr int MATOMS  = 512;   // atoms per molecule
constexpr int THREADS = 256;   // 8 wave32s
constexpr int WAVES   = THREADS / 32;
constexpr int TILES   = MATOMS / 16;          // 32 tiles of 16 atoms
constexpr int TPW     = TILES / WAVES;        // 4 tiles per wave

__global__ __launch_bounds__(THREADS)
void dihedral2coord_kernel(const float* __restrict__ input,          // (N, K)  theta
                           const float* __restrict__ pos0,           // (N, M, 3)
                           const int*   __restrict__ angles,         // (K, 4)
                           const unsigned char* __restrict__ mmask,  // (K, M) bool
                           float* __restrict__ out)                  // (N, M, 3)
{
    // Homogeneous-padded positions: (x, y, z, 1.0) per atom, 16B stride.
    // comp 3 is written once and never touched again -> B operand's "1" row
    // comes from a plain aligned ds_load_b64, no per-lane branching.
    __shared__ float sPos4[MATOMS * 4];   // 8 KB
    __shared__ float sTheta[KSTEPS];
    __shared__ int   sAng[KSTEPS * 4];

    const int n    = blockIdx.x;
    const int tid  = threadIdx.x;
    const int wave = tid >> 5;
    const int lane = tid & 31;
    const int col  = lane & 15;           // point-within-tile
    const int half = lane >> 4;           // 0: rows {0,1}=(x,y)  1: rows {2,3}=(z,1)

    // ---- stage molecule data into LDS (HBM touched once) ----
    const float* gpos = pos0 + (size_t)n * (MATOMS * 3);
    for (int i = tid; i < MATOMS; i += THREADS) {
        sPos4[4*i + 0] = gpos[3*i + 0];
        sPos4[4*i + 1] = gpos[3*i + 1];
        sPos4[4*i + 2] = gpos[3*i + 2];
        sPos4[4*i + 3] = 1.0f;
    }
    for (int i = tid; i < KSTEPS; i += THREADS)     sTheta[i] = input[(size_t)n * KSTEPS + i];
    for (int i = tid; i < KSTEPS * 4; i += THREADS) sAng[i] = angles[i];

    for (int k = 0; k < KSTEPS; ++k) {
        __syncthreads();  // previous step's writes (and initial fill) visible

        // ---- pivot atoms + dihedral angle (redundant per-thread; broadcast LDS) ----
        const int i0 = sAng[4 * k + 0];
        const int i1 = sAng[4 * k + 1];
        const int i2 = sAng[4 * k + 2];
        const int i3 = sAng[4 * k + 3];

        const float p0x = sPos4[4*i0+0], p0y = sPos4[4*i0+1], p0z = sPos4[4*i0+2];
        const float p1x = sPos4[4*i1+0], p1y = sPos4[4*i1+1], p1z = sPos4[4*i1+2];
        const float p2x = sPos4[4*i2+0], p2y = sPos4[4*i2+1], p2z = sPos4[4*i2+2];
        const float p3x = sPos4[4*i3+0], p3y = sPos4[4*i3+1], p3z = sPos4[4*i3+2];

        const float rIJx = p1x - p0x, rIJy = p1y - p0y, rIJz = p1z - p0z;
        const float rJKx = p2x - p1x, rJKy = p2y - p1y, rJKz = p2z - p1z;
        const float rKLx = p3x - p2x, rKLy = p3y - p2y, rKLz = p3z - p2z;

        // nIJK = rIJ x rJK ; nJKL = rJK x rKL ; m = nIJK x rJK
        const float nIx = rIJy*rJKz - rIJz*rJKy;
        const float nIy = rIJz*rJKx - rIJx*rJKz;
        const float nIz = rIJx*rJKy - rIJy*rJKx;
        const float nJx = rJKy*rKLz - rJKz*rKLy;
        const float nJy = rJKz*rKLx - rJKx*rKLz;
        const float nJz = rJKx*rKLy - rJKy*rKLx;
        const float mx  = nIy*rJKz - nIz*rJKy;
        const float my  = nIz*rJKx - nIx*rJKz;
        const float mz  = nIx*rJKy - nIy*rJKx;

        const float nIn = sqrtf(nIx*nIx + nIy*nIy + nIz*nIz);
        const float nJn = sqrtf(nJx*nJx + nJy*nJy + nJz*nJz);
        const float mn  = sqrtf(mx*mx + my*my + mz*mz);
        const float sin_arg = (mx*nJx + my*nJy + mz*nJz) / (nJn * mn);
        const float cos_arg = (nIx*nJx + nIy*nJy + nIz*nJz) / (nIn * nJn);
        const float value = sTheta[k] + atan2f(sin_arg, cos_arg);

        float s, c;
        sincosf(value, &s, &c);
        const float tt = 1.0f - c;

        // axis = normalize(p2 - p1), origin = p1
        float dn = sqrtf(rJKx*rJKx + rJKy*rJKy + rJKz*rJKz);
        dn = fmaxf(dn, 1e-12f);
        const float X = rJKx / dn, Y = rJKy / dn, Z = rJKz / dn;

        // Rodrigues rotation matrix
        const float R00 = tt*X*X + c;     const float R01 = tt*X*Y - s*Z;  const float R02 = tt*X*Z + s*Y;
        const float R10 = tt*Y*X + s*Z;   const float R11 = tt*Y*Y + c;    const float R12 = tt*Y*Z - s*X;
        const float R20 = tt*Z*X - s*Y;   const float R21 = tt*Z*Y + s*X;  const float R22 = tt*Z*Z + c;

        // translation so that rotated = R*p + t  ==  R*(p-begin)+begin
        const float tvx = p1x - (R00*p1x + R01*p1y + R02*p1z);
        const float tvy = p1y - (R10*p1x + R11*p1y + R12*p1z);
        const float tvz = p1z - (R20*p1x + R21*p1y + R22*p1z);

        // ---- build WMMA A operand: 16x4, rows 0..2 = [R | t], rest zero ----
        // lane L(<16): A[L][0], A[L][1]  ; lane L+16: A[L][2], A[L][3]
        v2f A; A.x = 0.0f; A.y = 0.0f;
        if (lane == 0)  { A.x = R00; A.y = R01; }
        if (lane == 1)  { A.x = R10; A.y = R11; }
        if (lane == 2)  { A.x = R20; A.y = R21; }
        if (lane == 16) { A.x = R02; A.y = tvx; }
        if (lane == 17) { A.x = R12; A.y = tvy; }
        if (lane == 18) { A.x = R22; A.y = tvz; }

        __syncthreads();  // everyone done reading pivots before any masked write

        const unsigned char* mrow = mmask + (size_t)k * MATOMS;

        // ---- each wave transforms its 4 tiles of 16 atoms via WMMA ----
        for (int ti = 0; ti < TPW; ++ti) {
            const int base = (wave * TPW + ti) << 4;
            const int pt   = base + col;

            // B operand: 4x16, column pt = (x, y, z, 1).
            // Uniform aligned 8B LDS load: half 0 -> comps {0,1}, half 1 -> {2,3}.
            const int boff = 4*pt + 2*half;
            v2f B;
            B.x = sPos4[boff + 0];
            B.y = sPos4[boff + 1];

            // Mask byte: unconditional (both halves read the same byte; L2-resident).
            const bool mv = mrow[pt] != 0;

            v8f Cz = {0.f,0.f,0.f,0.f,0.f,0.f,0.f,0.f};
            // D = A(16x4) x B(4x16): rows 0..2 of D are rotated x,y,z.
            // D row j -> VGPR j, lanes 0-15 => lane n holds point n's coords in D[0..2].
            v8f D = __builtin_amdgcn_wmma_f32_16x16x4_f32(
                false, A, false, B, (short)0, Cz, false, false);

            if ((lane < 16) && mv) {
                sPos4[4*pt + 0] = D[0];
                sPos4[4*pt + 1] = D[1];
                sPos4[4*pt + 2] = D[2];
                // comp 3 stays 1.0
            }
        }
    }

    __syncthreads();
    float* gout = out + (size_t)n * (MATOMS * 3);
    for (int i = tid; i < MATOMS; i += THREADS) {
        gout[3*i + 0] = sPos4[4*i + 0];
        gout[3*i + 1] = sPos4[4*i + 1];
        gout[3*i + 2] = sPos4[4*i + 2];
    }
}

extern "C" void kernel_launch(void* const* d_in, const int* in_sizes, int n_in,
                              void* d_out, int out_size, void* d_ws, size_t ws_size,
                              hipStream_t stream) {
    const float*         input  = (const float*)d_in[0];          // (N, K) f32
    const float*         pos0   = (const float*)d_in[1];          // (N, M, 3) f32
    const int*           angles = (const int*)d_in[2];            // (K, 4) i32
    const unsigned char* mmask  = (const unsigned char*)d_in[3];  // (K, M) bool
    float* out = (float*)d_out;                                   // (N, M, 3) f32

    const int N = in_sizes[0] / KSTEPS;  // 4096
    dihedral2coord_kernel<<<N, THREADS, 0, stream>>>(input, pos0, angles, mmask, out);
}